// Traveltime_34952443855268
// MI455X (gfx1250) — compile-verified
//
#include <hip/hip_runtime.h>
#include <hip/hip_bf16.h>

// Problem geometry (from reference): v[NT][NS][NR], argmax over time per trace.
#define NT      4000
#define NTRACE  8192                 // NS*NR = 16*512 independent traces
#define NQUAD   (NTRACE / 4)         // 2048: each thread owns 4 adjacent traces
#define NCHUNK  25                   // time chunks per input
#define STEPS   (NT / NCHUNK)        // 160 time steps per chunk (mult. of 8)
#define BLK     256
#define RING    8                    // LDS ring slots; RING-1 loads in flight

typedef __attribute__((ext_vector_type(4))) float v4f;
typedef __attribute__((ext_vector_type(4))) int   v4i;

// ---- CDNA5 async global->LDS copy (ASYNCcnt-tracked), non-temporal stream ----
__device__ __forceinline__ void async_copy_b128_nt(unsigned lds_byte_addr,
                                                   const void* gptr) {
  asm volatile("global_load_async_to_lds_b128 %0, %1, off th:TH_LOAD_NT"
               :: "v"(lds_byte_addr), "v"(gptr)
               : "memory");
}
__device__ __forceinline__ void wait_async_le7() {
  asm volatile("s_wait_asynccnt 0x7" ::: "memory");
}
__device__ __forceinline__ void wait_async_0() {
  asm volatile("s_wait_asynccnt 0x0" ::: "memory");
}

// ---------------------------------------------------------------------------
// Stage 1: stream one time-chunk of one input, per-trace running argmax(v^2).
// grid = (NQUAD/BLK, NCHUNK, 2), block = 256 (8 wave32).
// LDS ring: 8 slots x 256 threads x 16B = 32 KB; 7 async loads in flight
// (512 B per wave per op -> 3.5 KB/wave in flight; x ~3200 waves ~= 11 MB,
// sized against the ~14 MB bandwidth-delay product of 23.3 TB/s HBM).
// ---------------------------------------------------------------------------
__global__ __launch_bounds__(BLK) void tt_stage1(
    const float* __restrict__ x, const float* __restrict__ y,
    float* __restrict__ wmax, int* __restrict__ widx) {
  __shared__ v4f lbuf[RING][BLK];                    // 32 KB ring
  const int tid   = threadIdx.x;
  const int q     = blockIdx.x * BLK + tid;          // quad-trace id [0, NQUAD)
  const int chunk = blockIdx.y;
  const int z     = blockIdx.z;                      // 0 = x, 1 = y
  const float* src = z ? y : x;
  const int t0 = chunk * STEPS;

  const float* g = src + (size_t)t0 * NTRACE + (size_t)q * 4;

  // LDS byte address of this thread's slot 0 (low 32 bits of the generic
  // pointer are the LDS offset per the CDNA5 aperture layout).
  // Slot stride = BLK * 16 B = 4 KB.
  const unsigned myoff = (unsigned)(size_t)&lbuf[0][tid];

  // Prologue: RING-1 = 7 loads in flight.
#pragma unroll
  for (unsigned p = 0; p < RING - 1; ++p)
    async_copy_b128_nt(myoff + (p << 12), g + (size_t)p * NTRACE);
  const float* gpre = g + (size_t)(RING - 1) * NTRACE;  // next address to issue

  float m0 = -1.f, m1 = -1.f, m2 = -1.f, m3 = -1.f;  // v^2 >= 0, so -1 < all
  int   i0 = t0,  i1 = t0,  i2 = t0,  i3 = t0;

#pragma unroll 8
  for (int i = 0; i < STEPS; ++i) {
    const int n = i + (RING - 1);
    if (n < STEPS) {
      // Issue into slot (i+7)&7: its previous payload (load i-1) was
      // consumed last iteration. After issue, <=8 outstanding; in-order
      // completion => wait <=7 guarantees load i has landed in LDS.
      async_copy_b128_nt(myoff + ((unsigned)(n & (RING - 1)) << 12), gpre);
      gpre += NTRACE;
      wait_async_le7();
    } else {
      wait_async_0();                                 // tail flush (last 7)
    }
    const v4f v = lbuf[i & (RING - 1)][tid];          // ds_load_b128
    const int t = t0 + i;
    const float e0 = v.x * v.x, e1 = v.y * v.y;
    const float e2 = v.z * v.z, e3 = v.w * v.w;
    if (e0 > m0) { m0 = e0; i0 = t; }                 // strict > : first-
    if (e1 > m1) { m1 = e1; i1 = t; }                 // occurrence tie rule,
    if (e2 > m2) { m2 = e2; i2 = t; }                 // matching jnp.argmax
    if (e3 > m3) { m3 = e3; i3 = t; }
  }

  const size_t o = ((size_t)(z * NCHUNK + chunk)) * NTRACE + (size_t)q * 4;
  v4f mv; mv.x = m0; mv.y = m1; mv.z = m2; mv.w = m3;
  v4i iv; iv.x = i0; iv.y = i1; iv.z = i2; iv.w = i3;
  *(v4f*)(wmax + o) = mv;
  *(v4i*)(widx + o) = iv;
}

// ---------------------------------------------------------------------------
// Stage 2: combine chunk partials (increasing time => strict > keeps the
// first occurrence), emit per-trace squared index difference.
// ---------------------------------------------------------------------------
__global__ __launch_bounds__(BLK) void tt_stage2(
    const float* __restrict__ wmax, const int* __restrict__ widx,
    float* __restrict__ wdsq) {
  const int j = blockIdx.x * BLK + threadIdx.x;       // trace id [0, NTRACE)
  int bi[2];
#pragma unroll
  for (int z = 0; z < 2; ++z) {
    float best = -2.f;
    int bidx = 0;
    for (int c = 0; c < NCHUNK; ++c) {
      const size_t o = ((size_t)(z * NCHUNK + c)) * NTRACE + j;
      const float m = wmax[o];
      const int  ii = widx[o];
      if (m > best) { best = m; bidx = ii; }
    }
    bi[z] = bidx;
  }
  const float d = (float)(bi[0] - bi[1]);             // |d| <= 3999: exact f32
  wdsq[j] = d * d;
}

// ---------------------------------------------------------------------------
// Stage 3: deterministic tree reduction of 8192 values -> mean.
// ---------------------------------------------------------------------------
__global__ __launch_bounds__(BLK) void tt_stage3(
    const float* __restrict__ wdsq, float* __restrict__ out) {
  __shared__ float red[BLK];
  float s = 0.f;
  for (int k = threadIdx.x; k < NTRACE; k += BLK) s += wdsq[k];
  red[threadIdx.x] = s;
  __syncthreads();
  for (int w = BLK / 2; w > 0; w >>= 1) {
    if ((int)threadIdx.x < w) red[threadIdx.x] += red[threadIdx.x + w];
    __syncthreads();
  }
  if (threadIdx.x == 0) out[0] = red[0] * (1.0f / (float)NTRACE);
}

extern "C" void kernel_launch(void* const* d_in, const int* in_sizes, int n_in,
                              void* d_out, int out_size, void* d_ws, size_t ws_size,
                              hipStream_t stream) {
  (void)in_sizes; (void)n_in; (void)out_size; (void)ws_size;
  const float* x = (const float*)d_in[0];
  const float* y = (const float*)d_in[1];
  float* out = (float*)d_out;

  char* ws = (char*)d_ws;
  const size_t PARTIALS = (size_t)2 * NCHUNK * NTRACE;      // 409,600 elems
  float* wmax = (float*)ws;                                 // 1.64 MB
  int*   widx = (int*)(ws + PARTIALS * sizeof(float));      // 1.64 MB
  float* wdsq = (float*)(ws + 2 * PARTIALS * sizeof(float));// 32 KB

  dim3 g1(NQUAD / BLK, NCHUNK, 2);                          // (8, 25, 2)
  tt_stage1<<<g1, BLK, 0, stream>>>(x, y, wmax, widx);
  tt_stage2<<<NTRACE / BLK, BLK, 0, stream>>>(wmax, widx, wdsq);
  tt_stage3<<<1, BLK, 0, stream>>>(wdsq, out);
}